// SegmentScore_83322365542780
// MI455X (gfx1250) — compile-verified
//
#include <hip/hip_runtime.h>
#include <math.h>

#define BB 16
#define FF 128
#define HH 144
#define EE 24
#define CC0 12

typedef __attribute__((ext_vector_type(2))) float v2f;
typedef __attribute__((ext_vector_type(4))) float v4f;
typedef __attribute__((ext_vector_type(8))) float v8f;

// ---------------------------------------------------------------------------
// Kernel 1: HarmonyEncoder.  One thread per (b,h) row: t = [hv[:12]@W0.T+b0,
// hv[12:]@W1.T+b1], q = t@Wc.T+bc.  Tiny (2304 threads).
// ---------------------------------------------------------------------------
__global__ void harmony_encode(const float* __restrict__ hv,
                               const float* __restrict__ W0, const float* __restrict__ b0,
                               const float* __restrict__ W1, const float* __restrict__ b1,
                               const float* __restrict__ Wc, const float* __restrict__ bc,
                               float* __restrict__ q) {
  int idx = blockIdx.x * blockDim.x + threadIdx.x;
  if (idx >= BB * HH) return;
  const float* v = hv + (size_t)idx * EE;
  float t[EE];
#pragma unroll
  for (int j = 0; j < CC0; ++j) {
    float s0 = b0[j], s1 = b1[j];
#pragma unroll
    for (int k = 0; k < CC0; ++k) {
      s0 += v[k]       * W0[j * CC0 + k];
      s1 += v[CC0 + k] * W1[j * CC0 + k];
    }
    t[j] = s0;
    t[CC0 + j] = s1;
  }
  float* qo = q + (size_t)idx * EE;
#pragma unroll
  for (int e = 0; e < EE; ++e) {
    float s = bc[e];
#pragma unroll
    for (int j = 0; j < EE; ++j) s += Wc[e * EE + j] * t[j];
    qo[e] = s;
  }
}

// ---------------------------------------------------------------------------
// Kernel 2: d[b,h,l] = sum_e q[b,h,e]*note[b,l,e] via V_WMMA_F32_16X16X4_F32.
// One wave per 16x16 output tile; E=24 -> 6 chained WMMAs (K=4 each).
// A(16x4): lanes 0-15 hold M with K={0,1}, lanes 16-31 hold M with K={2,3}.
// B(4x16): lanes 0-15 hold N with K={0,1}, lanes 16-31 hold N with K={2,3}.
// C/D(16x16): VGPR r -> row r (+8 for lanes 16-31), col = lane&15.
// Grid sized so every wave is fully active (EXEC all-ones as WMMA requires).
// ---------------------------------------------------------------------------
__global__ void dots_wmma(const float* __restrict__ q,
                          const float* __restrict__ notes,
                          float* __restrict__ d) {
  const int HT = HH / 16;          // 9
  const int FT = FF / 16;          // 8
  int wave = (int)((blockIdx.x * blockDim.x + threadIdx.x) >> 5);
  int lane = threadIdx.x & 31;
  if (wave >= BB * HT * FT) return;        // uniform per wave (never taken: exact grid)
  int ft = wave % FT;
  int ht = (wave / FT) % HT;
  int b  = wave / (FT * HT);
  int h0 = ht * 16, f0 = ft * 16;

  int mn    = lane & 15;                   // row (A) / col (B,D)
  int khalf = (lane >> 4) << 1;            // 0 or 2

  const float* qb = q     + (size_t)b * HH * EE;
  const float* nb = notes + (size_t)b * FF * EE;

  v8f c = {};
#pragma unroll
  for (int e0 = 0; e0 < EE; e0 += 4) {
    v2f a, bmat;
    a.x    = qb[(h0 + mn) * EE + e0 + khalf];
    a.y    = qb[(h0 + mn) * EE + e0 + khalf + 1];
    bmat.x = nb[(f0 + mn) * EE + e0 + khalf];
    bmat.y = nb[(f0 + mn) * EE + e0 + khalf + 1];
    c = __builtin_amdgcn_wmma_f32_16x16x4_f32(
        /*neg_a=*/false, a, /*neg_b=*/false, bmat,
        /*c_mod=*/(short)0, c, /*reuse_a=*/false, /*reuse_b=*/false);
  }

  float* dout = d + (size_t)b * HH * FF;
  int row_off = (lane >> 4) * 8;
#pragma unroll
  for (int r = 0; r < 8; ++r)
    dout[(size_t)(h0 + r + row_off) * FF + f0 + mn] = c[r];
}

// ---------------------------------------------------------------------------
// Kernel 3: per-(b,h) row max, exp, and cumulative sums over F, written
// TRANSPOSED as [b][f][h] so kernel 4's reads/writes coalesce/vectorize on h.
// ---------------------------------------------------------------------------
__global__ void scan_rows(const float* __restrict__ d,
                          float* __restrict__ numT,
                          float* __restrict__ denT) {
  int idx = blockIdx.x * blockDim.x + threadIdx.x;
  if (idx >= BB * HH) return;
  int h = idx % HH;
  int b = idx / HH;
  const float* row = d + (size_t)idx * FF;
  const float scale = 0.2041241452319315f;   // 1/sqrt(24)

  float m = -INFINITY;
  for (int l = 0; l < FF; ++l) m = fmaxf(m, row[l] * scale);

  float num = 0.f, den = 0.f;
  float* nT = numT + (size_t)b * FF * HH + h;
  float* dT = denT + (size_t)b * FF * HH + h;
  for (int l = 0; l < FF; ++l) {
    float dv = row[l];
    float p  = expf(dv * scale - m);
    num += p * dv;
    den += p;
    nT[(size_t)l * HH] = num;
    dT[(size_t)l * HH] = den;
  }
}

// ---------------------------------------------------------------------------
// Kernel 4: the 151 MB streaming stage (bandwidth bound).  One thread per
// FOUR consecutive h of one [b][s][e] cell: b128 loads of numT/denT (hot,
// cached) and one non-temporal global_store_b128 for the write-once output
// so it doesn't rinse the 2.4 MB num/den working set out of WGP$/L2.
// ---------------------------------------------------------------------------
__global__ void segment_scores(const float* __restrict__ numT,
                               const float* __restrict__ denT,
                               float* __restrict__ out) {
  const int HQ = HH / 4;                                   // 36
  size_t idx = (size_t)blockIdx.x * blockDim.x + threadIdx.x;
  const size_t N = (size_t)BB * FF * FF * HQ;
  if (idx >= N) return;
  int h = (int)(idx % HQ) * 4;
  size_t t = idx / HQ;
  int e = (int)(t % FF); t /= FF;
  int s = (int)(t % FF);
  int b = (int)(t / FF);

  v4f sc = {0.f, 0.f, 0.f, 0.f};
  if (e >= s) {
    const float* nbase = numT + (size_t)b * FF * HH + h;
    const float* dbase = denT + (size_t)b * FF * HH + h;
    v4f n_e = *(const v4f*)(nbase + (size_t)e * HH);
    v4f d_e = *(const v4f*)(dbase + (size_t)e * HH);
    v4f n_s = {0.f, 0.f, 0.f, 0.f};
    v4f d_s = {0.f, 0.f, 0.f, 0.f};
    if (s > 0) {
      n_s = *(const v4f*)(nbase + (size_t)(s - 1) * HH);
      d_s = *(const v4f*)(dbase + (size_t)(s - 1) * HH);
    }
    sc = (n_e - n_s) / (d_e - d_s);
  }
  __builtin_nontemporal_store(sc, (v4f*)(out + idx * 4));
}

// ---------------------------------------------------------------------------
extern "C" void kernel_launch(void* const* d_in, const int* in_sizes, int n_in,
                              void* d_out, int out_size, void* d_ws, size_t ws_size,
                              hipStream_t stream) {
  const float* notes = (const float*)d_in[0];  // [B,F,E]
  const float* hv    = (const float*)d_in[1];  // [B,H,E]
  const float* W0    = (const float*)d_in[2];
  const float* b0    = (const float*)d_in[3];
  const float* W1    = (const float*)d_in[4];
  const float* b1    = (const float*)d_in[5];
  const float* Wc    = (const float*)d_in[6];
  const float* bc    = (const float*)d_in[7];

  float* ws   = (float*)d_ws;
  float* q    = ws;                            // B*H*E = 55296
  float* dbuf = q    + (size_t)BB * HH * EE;   // B*H*F = 294912
  float* numT = dbuf + (size_t)BB * HH * FF;   // B*F*H = 294912
  float* denT = numT + (size_t)BB * FF * HH;   // B*F*H = 294912

  harmony_encode<<<(BB * HH + 127) / 128, 128, 0, stream>>>(hv, W0, b0, W1, b1, Wc, bc, q);

  int waves = BB * (HH / 16) * (FF / 16);  // 1152 tiles, 8 waves/block -> 144 blocks exact
  dots_wmma<<<waves / 8, 256, 0, stream>>>(q, notes, dbuf);

  scan_rows<<<(BB * HH + 127) / 128, 128, 0, stream>>>(dbuf, numT, denT);

  size_t Nq = (size_t)BB * FF * FF * (HH / 4);   // 9,437,184 threads (x4 h each)
  segment_scores<<<(unsigned)((Nq + 255) / 256), 256, 0, stream>>>(numT, denT, (float*)d_out);
}